// Model_85272280695019
// MI455X (gfx1250) — compile-verified
//
#include <hip/hip_runtime.h>
#include <hip/hip_bf16.h>

#define SLOPE 0.01f

typedef __attribute__((ext_vector_type(2))) float v2f;
typedef __attribute__((ext_vector_type(8))) float v8f;

__device__ __forceinline__ float lrelu(float x) {
    return x >= 0.0f ? x : SLOPE * x;
}

// Monotone order-preserving map float -> uint so unsigned atomicMax == float max.
__device__ __forceinline__ unsigned enc_f32(float f) {
    unsigned u = __float_as_uint(f);
    return (u >> 31) ? ~u : (u | 0x80000000u);
}
__device__ __forceinline__ float dec_f32(unsigned u) {
    return __uint_as_float((u >> 31) ? (u & 0x7FFFFFFFu) : ~u);
}

// ---------------- kernel 0: zero workspace -------------------------------
__global__ void k_zero(float* __restrict__ ws, size_t n) {
    size_t i = (size_t)blockIdx.x * blockDim.x + threadIdx.x;
    size_t stride = (size_t)gridDim.x * blockDim.x;
    for (; i < n; i += stride) ws[i] = 0.0f;
}

// ---------------- kernel 1: per-node attention projections ---------------
// One wave32 per node: ps = h . Wa[:D], pd = h . Wa[D:]
__global__ void k_proj(const float* __restrict__ node_h,
                       const float* __restrict__ W_attn,
                       float* __restrict__ ps, float* __restrict__ pd,
                       int N, int D) {
    int lane = threadIdx.x & 31;
    int wave = threadIdx.x >> 5;
    int node = blockIdx.x * (blockDim.x >> 5) + wave;
    if (node >= N) return;
    const float* h = node_h + (size_t)node * D;
    float s0 = 0.0f, s1 = 0.0f;
    for (int i = lane; i < D; i += 32) {
        float hv = h[i];
        s0 += hv * W_attn[i];
        s1 += hv * W_attn[D + i];
    }
    #pragma unroll
    for (int off = 16; off > 0; off >>= 1) {
        s0 += __shfl_down(s0, off, 32);
        s1 += __shfl_down(s1, off, 32);
    }
    if (lane == 0) { ps[node] = s0; pd[node] = s1; }
}

// ---------------- kernel 2: edge scores, degree, segment max -------------
__global__ void k_edge_score(const float* __restrict__ ps,
                             const float* __restrict__ pd,
                             const float* __restrict__ b_attn,
                             const int* __restrict__ src,
                             const int* __restrict__ dst,
                             float* __restrict__ w,
                             unsigned* __restrict__ deg,
                             unsigned* __restrict__ wmax_enc,
                             int E) {
    int e = blockIdx.x * blockDim.x + threadIdx.x;
    if (e >= E) return;
    int s = src[e], d = dst[e];
    float wv = lrelu(ps[s] + pd[d] + b_attn[0]);
    w[e] = wv;
    atomicAdd(&deg[d], 1u);
    atomicMax(&wmax_enc[d], enc_f32(wv));
}

// ---------------- kernel 3: exp and denominator --------------------------
__global__ void k_edge_exp(const float* __restrict__ w,
                           const int* __restrict__ dst,
                           const unsigned* __restrict__ wmax_enc,
                           float* __restrict__ ew,
                           float* __restrict__ denom,
                           int E) {
    int e = blockIdx.x * blockDim.x + threadIdx.x;
    if (e >= E) return;
    int d = dst[e];
    float ewv = __expf(w[e] - dec_f32(wmax_enc[d]));
    ew[e] = ewv;
    atomicAdd(&denom[d], ewv);
}

// ---------------- kernel 4: weighted message scatter-max -----------------
// One wave32 per edge; lanes stride D. agg_enc holds monotone-encoded floats.
__global__ void k_edge_scatter(const float* __restrict__ ew,
                               const float* __restrict__ denom,
                               const float* __restrict__ node_h,
                               const int* __restrict__ src,
                               const int* __restrict__ dst,
                               unsigned* __restrict__ agg_enc,
                               int E, int D) {
    int lane = threadIdx.x & 31;
    int wave = threadIdx.x >> 5;
    int e = blockIdx.x * (blockDim.x >> 5) + wave;
    if (e >= E) return;
    int s = src[e], d = dst[e];
    float dn = denom[d];
    float a = ew[e] / (dn > 0.0f ? dn : 1.0f);
    const float* hs = node_h + (size_t)s * D;
    unsigned* row = agg_enc + (size_t)d * D;
    for (int i = lane; i < D; i += 32) {
        atomicMax(&row[i], enc_f32(a * hs[i]));
    }
}

// ---------------- kernel 5: gated update + per-graph sum -----------------
__global__ void k_node_update(const float* __restrict__ node_h,
                              const float* __restrict__ eta,
                              const unsigned* __restrict__ deg,
                              const unsigned* __restrict__ agg_enc,
                              const int* __restrict__ graph_ids,
                              float* __restrict__ hg,
                              int N, int D) {
    int lane = threadIdx.x & 31;
    int wave = threadIdx.x >> 5;
    int node = blockIdx.x * (blockDim.x >> 5) + wave;
    if (node >= N) return;
    int g = graph_ids[node];
    unsigned dg = deg[node];
    float et = eta[node];
    const float* h = node_h + (size_t)node * D;
    const unsigned* ar = agg_enc + (size_t)node * D;
    float* hrow = hg + (size_t)g * D;
    for (int i = lane; i < D; i += 32) {
        float hv = h[i];
        float hn;
        if (dg > 0u) {
            float ag = dec_f32(ar[i]);
            hn = et * hv + (1.0f - et) * ag;
        } else {
            hn = hv;
        }
        atomicAdd(&hrow[i], hn);
    }
}

// ---------------- kernel 6: classifier GEMM via fp32 WMMA ----------------
// out[B,C] = lrelu(hg)[B,D] @ W_lin[D,C] + b_lin.
// V_WMMA_F32_16X16X4_F32.  A 16x4: lanes 0-15 hold M=lane K={0,1},
// lanes 16-31 hold M=lane-16 K={2,3}.  B 4x16 mirrors (K rows across the
// lane halves).  C/D 16x16: vgpr r -> row r (lanes 0-15) / r+8 (lanes 16-31).
//
// Out-of-range rows/cols are handled by CLAMPED addressing on the loads
// (valid duplicate data, never stored) so the inner loop has no divergence
// and no exec-mask save/restore around the WMMA.
__global__ void k_classifier(const float* __restrict__ hg,
                             const float* __restrict__ W_lin,
                             const float* __restrict__ b_lin,
                             float* __restrict__ out,
                             int B, int D, int C) {
    int lane  = threadIdx.x & 31;
    int ntile = threadIdx.x >> 5;          // wave 0: cols 0-15, wave 1: cols 16-31
    int mtile = blockIdx.x;

    int m  = lane & 15;
    int kq = (lane >> 4) * 2;              // 0 or 2 within the K=4 slab

    int row  = mtile * 16 + m;
    int col  = ntile * 16 + (lane & 15);
    int rowL = row < B ? row : B - 1;      // clamp: duplicate data, discarded at store
    int colL = col < C ? col : C - 1;

    const float* hr = hg + (size_t)rowL * D + kq;          // A: contiguous pair
    const float* wl = W_lin + (size_t)kq * C + colL;       // B: stride-C pair

    v8f acc = {};
    for (int k = 0; k < D; k += 4) {
        v2f a, b;
        a.x = lrelu(hr[k]);
        a.y = lrelu(hr[k + 1]);
        b.x = wl[(size_t)k * C];
        b.y = wl[(size_t)(k + 1) * C];
        acc = __builtin_amdgcn_wmma_f32_16x16x4_f32(
            /*neg_a=*/false, a, /*neg_b=*/false, b,
            /*c_mod=*/(short)0, acc, /*reuse_a=*/false, /*reuse_b=*/false);
    }

    if (col < C) {
        float bias = b_lin[col];
        int base = mtile * 16 + (lane >> 4) * 8;
        #pragma unroll
        for (int r = 0; r < 8; ++r) {
            int rr = base + r;
            if (rr < B) out[(size_t)rr * C + col] = acc[r] + bias;
        }
    }
}

extern "C" void kernel_launch(void* const* d_in, const int* in_sizes, int n_in,
                              void* d_out, int out_size, void* d_ws, size_t ws_size,
                              hipStream_t stream) {
    const float* node_h    = (const float*)d_in[0];
    const float* eta       = (const float*)d_in[1];
    const float* W_attn    = (const float*)d_in[2];
    const float* b_attn    = (const float*)d_in[3];
    const float* W_lin     = (const float*)d_in[4];
    const float* b_lin     = (const float*)d_in[5];
    const int*   src       = (const int*)d_in[6];
    const int*   dst       = (const int*)d_in[7];
    const int*   graph_ids = (const int*)d_in[8];

    const int N = in_sizes[1];               // eta is [N,1]
    const int D = in_sizes[0] / N;           // node_h is [N,D]
    const int E = in_sizes[6];               // src is [E]
    const int C = in_sizes[5];               // b_lin is [C]
    const int B = out_size / C;              // out is [B,C]

    float* ws = (float*)d_ws;
    size_t o = 0;
    float*    ps       = ws + o; o += (size_t)N;
    float*    pd       = ws + o; o += (size_t)N;
    unsigned* deg      = (unsigned*)(ws + o); o += (size_t)N;
    unsigned* wmax_enc = (unsigned*)(ws + o); o += (size_t)N;
    float*    denom    = ws + o; o += (size_t)N;
    float*    w        = ws + o; o += (size_t)E;
    float*    ew       = ws + o; o += (size_t)E;
    unsigned* agg_enc  = (unsigned*)(ws + o); o += (size_t)N * D;
    float*    hg       = ws + o; o += (size_t)B * D;
    const size_t total_ws_elems = o;

    const int TB = 256;               // 8 waves per block
    const int wavesPerBlock = TB / 32;

    // 0) zero the whole workspace (deg/wmax/denom/agg/hg need it).
    {
        size_t nb = (total_ws_elems + TB - 1) / TB;
        if (nb > 65535u * 16u) nb = 65535u * 16u;
        k_zero<<<dim3((unsigned)nb), dim3(TB), 0, stream>>>(ws, total_ws_elems);
    }
    // 1) per-node projections
    k_proj<<<dim3((N + wavesPerBlock - 1) / wavesPerBlock), dim3(TB), 0, stream>>>(
        node_h, W_attn, ps, pd, N, D);
    // 2) edge scores + degree + segment max
    k_edge_score<<<dim3((E + TB - 1) / TB), dim3(TB), 0, stream>>>(
        ps, pd, b_attn, src, dst, w, deg, wmax_enc, E);
    // 3) exp + denominator
    k_edge_exp<<<dim3((E + TB - 1) / TB), dim3(TB), 0, stream>>>(
        w, dst, wmax_enc, ew, denom, E);
    // 4) weighted message scatter-max (wave per edge)
    k_edge_scatter<<<dim3((E + wavesPerBlock - 1) / wavesPerBlock), dim3(TB), 0, stream>>>(
        ew, denom, node_h, src, dst, agg_enc, E, D);
    // 5) gated update + per-graph sum (wave per node)
    k_node_update<<<dim3((N + wavesPerBlock - 1) / wavesPerBlock), dim3(TB), 0, stream>>>(
        node_h, eta, deg, agg_enc, graph_ids, hg, N, D);
    // 6) classifier GEMM with fp32 WMMA: 2 waves per block (one per 16-col tile)
    k_classifier<<<dim3((B + 15) / 16), dim3(64), 0, stream>>>(
        hg, W_lin, b_lin, (float*)d_out, B, D, C);
}